// UnitrackCriterion_11020886082192
// MI455X (gfx1250) — compile-verified
//
#include <hip/hip_runtime.h>
#include <math.h>

// ---------------------------------------------------------------------------
// UnitrackCriterion for MI455X (gfx1250, wave32).
// B=16 batches, N=2048 preds, M=1024 gts, T=256 track ids.
// Latency-bound problem (~0.8MB data): per-batch LDS staging, LDS float
// atomics for per-track histograms, and V_WMMA_F32_16X16X4_F32 ones-matmul
// used as the 256-element f32 block reducer (exact f32 accumulate).
// ---------------------------------------------------------------------------

#define BATCH 16
#define NPRED 2048
#define MGT   1024
#define TIDS  256

#define IOU_THR_C 0.5f
#define A_TRK 2.0f
#define A_SPA 1.5f
#define A_TMP 1.8f
#define B_FP  0.9f
#define B_FN  0.9f
#define G_SW  1.5f

// Per-batch workspace layout (floats):
// [0]=fn  [1]=fp  [2]=spatial(gated,normed)  [3]=n_uni  [4]=n_valid
// [5]=match(atomic)  [6]=switch(atomic)  [7]=temporal
// [8..263]   gt_cnt[256]
// [264..519] pr_cnt[256]
#define WS_SCAL 8
#define WS_GT   WS_SCAL
#define WS_PR   (WS_SCAL + TIDS)
#define WS_PER_B (WS_SCAL + 2 * TIDS)   // 520 floats/batch -> ~33KB total

__device__ __forceinline__ float scale_const() {
  // 1 / image diagonal; constant-folded float sqrt.
  return 1.0f / sqrtf(1920.0f * 1920.0f + 1080.0f * 1080.0f);
}

typedef float v2f __attribute__((ext_vector_type(2)));
typedef float v8f __attribute__((ext_vector_type(8)));

// Sum 256 f32 values from LDS using 4 chained V_WMMA_F32_16X16X4_F32 ops
// (A = data tile 16x4, B = ones 4x16, C = running accumulator).
// Must be executed by one full wave32 (threads 0..31, EXEC all ones).
// Returns the total on every lane of the wave.
__device__ __forceinline__ float wmma_reduce256(const float* s) {
  const int l = threadIdx.x & 31;
  const int m = l & 15;                    // matrix row handled by this lane
  const int kbase = (l < 16) ? 0 : 2;      // ISA A-layout: lanes16-31 hold K=2,3
  v2f bones;
  bones[0] = 1.0f;
  bones[1] = 1.0f;
  v8f acc = {};
#pragma unroll
  for (int c = 0; c < 4; ++c) {
    v2f a;
    a[0] = s[c * 64 + m * 4 + kbase + 0];
    a[1] = s[c * 64 + m * 4 + kbase + 1];
    // D[m,n] = sum_k A[m,k] * 1 + C[m,n]  -> chained row-sum accumulation
    acc = __builtin_amdgcn_wmma_f32_16x16x4_f32(
        /*neg_a=*/false, a, /*neg_b=*/false, bones,
        /*c_mod=*/(short)0, acc, /*reuse_a=*/false, /*reuse_b=*/false);
  }
  // Lanes 0-15: D vgpr r holds M=r ; lanes 16-31: M=8+r (same value for all N).
  float t = acc[0] + acc[1] + acc[2] + acc[3] + acc[4] + acc[5] + acc[6] + acc[7];
  t += __shfl_xor(t, 16, 32);              // combine the two M-halves
  return t;
}

// ---------------------------------------------------------------------------
// Kernel 1: per-batch per-track histograms + scalar stats.
// ---------------------------------------------------------------------------
__global__ void stats_kernel(const float* __restrict__ pb, const float* __restrict__ gb,
                             const int* __restrict__ pid, const int* __restrict__ gid,
                             float* __restrict__ ws) {
  const int b = blockIdx.x;
  const int tid = threadIdx.x;
  __shared__ float s_gt[TIDS], s_pr[TIDS], s_sw[TIDS], s_sh[TIDS];
  __shared__ float s_dw[TIDS], s_dh[TIDS], s_red[TIDS];

  s_gt[tid] = 0.0f; s_pr[tid] = 0.0f; s_sw[tid] = 0.0f;
  s_sh[tid] = 0.0f; s_dw[tid] = 0.0f; s_dh[tid] = 0.0f;
  __syncthreads();

  for (int j = tid; j < MGT; j += blockDim.x)
    atomicAdd(&s_gt[gid[b * MGT + j]], 1.0f);

  const float4* pb4 = (const float4*)pb;
  for (int i = tid; i < NPRED; i += blockDim.x) {
    int t = pid[b * NPRED + i];
    if (t > 0) {
      float4 p = pb4[b * NPRED + i];
      atomicAdd(&s_pr[t], 1.0f);
      atomicAdd(&s_sw[t], p.z - p.x);
      atomicAdd(&s_sh[t], p.w - p.y);
    }
  }
  __syncthreads();

  for (int i = tid; i < NPRED; i += blockDim.x) {
    int t = pid[b * NPRED + i];
    if (t > 0) {
      float4 p = pb4[b * NPRED + i];
      float cs = fmaxf(s_pr[t], 1.0f);
      atomicAdd(&s_dw[t], fabsf((p.z - p.x) - s_sw[t] / cs));
      atomicAdd(&s_dh[t], fabsf((p.w - p.y) - s_sh[t] / cs));
    }
  }
  __syncthreads();

  // Export per-track counts for the IoU kernel.
  ws[b * WS_PER_B + WS_GT + tid] = s_gt[tid];
  ws[b * WS_PER_B + WS_PR + tid] = s_pr[tid];

  // Five per-track reductions via WMMA ones-matmul.
  float r_fn = 0.0f, r_fp = 0.0f, r_nv = 0.0f, r_nu = 0.0f, r_sp = 0.0f;
  for (int st = 0; st < 5; ++st) {
    float g = s_gt[tid];
    float p = s_pr[tid];
    float term;
    if (st == 0)      term = (p == 0.0f) ? g : 0.0f;                       // fn
    else if (st == 1) term = (g == 0.0f) ? p : 0.0f;                       // fp
    else if (st == 2) term = p;                                            // n_valid
    else if (st == 3) term = (p > 0.0f) ? 1.0f : 0.0f;                     // n_uni
    else              term = (p > 1.0f) ? (s_dw[tid] + s_dh[tid]) * scale_const() / p
                                        : 0.0f;                            // spatial
    s_red[tid] = term;
    __syncthreads();
    if (tid < 32) {
      float tot = wmma_reduce256(s_red);
      if (st == 0)      r_fn = tot;
      else if (st == 1) r_fp = tot;
      else if (st == 2) r_nv = tot;
      else if (st == 3) r_nu = tot;
      else              r_sp = tot;
    }
    __syncthreads();
  }

  if (tid == 0) {
    float* w = ws + b * WS_PER_B;
    w[0] = r_fn;
    w[1] = r_fp;
    w[2] = (r_nv > 1.0f) ? r_sp / fmaxf(r_nu, 1.0f) : 0.0f;
    w[3] = r_nu;
    w[4] = r_nv;
    w[5] = 0.0f;   // match accumulator (kernel 2 atomics)
    w[6] = 0.0f;   // switch accumulator (kernel 2 atomics)
    w[7] = 0.0f;   // temporal (kernel 3)
  }
}

// ---------------------------------------------------------------------------
// Kernel 2: N x M IoU sweep -> match loss + switch count.
// Grid: (NPRED/256, BATCH). Each thread owns one pred, sweeps gts from LDS.
// ---------------------------------------------------------------------------
__global__ void iou_kernel(const float* __restrict__ pb, const float* __restrict__ gb,
                           const int* __restrict__ pid, const int* __restrict__ gid,
                           float* __restrict__ ws) {
  const int b = blockIdx.y;
  const int tid = threadIdx.x;
  __shared__ float4 sgb[MGT];
  __shared__ float sa2[MGT];
  __shared__ int   sgid[MGT];
  __shared__ float sprc[TIDS], sgtc[TIDS], s_red[TIDS];

  const float4* gb4 = (const float4*)gb;
  for (int j = tid; j < MGT; j += blockDim.x) {
    float4 g = gb4[b * MGT + j];
    sgb[j] = g;
    sa2[j] = fmaxf(g.z - g.x, 0.0f) * fmaxf(g.w - g.y, 0.0f);
    sgid[j] = gid[b * MGT + j];
  }
  sprc[tid] = ws[b * WS_PER_B + WS_PR + tid];
  sgtc[tid] = ws[b * WS_PER_B + WS_GT + tid];
  __syncthreads();

  const int i = blockIdx.x * blockDim.x + tid;
  const float4* pb4 = (const float4*)pb;
  __builtin_prefetch(&pb4[b * NPRED + i], 0, 3);   // global_prefetch_b8
  float4 p = pb4[b * NPRED + i];
  int t = pid[b * NPRED + i];
  bool valid = (t > 0);
  float a1 = fmaxf(p.z - p.x, 0.0f) * fmaxf(p.w - p.y, 0.0f);

  float maxiou = -1e9f;
  float sw = 0.0f;
  for (int j = 0; j < MGT; ++j) {
    float4 g = sgb[j];
    float lx = fmaxf(p.x, g.x), ly = fmaxf(p.y, g.y);
    float rx = fminf(p.z, g.z), ry = fminf(p.w, g.w);
    float inter = fmaxf(rx - lx, 0.0f) * fmaxf(ry - ly, 0.0f);
    float uni = a1 + sa2[j] - inter;
    float iou = inter / fmaxf(uni, 1e-6f);
    int gj = sgid[j];
    if (gj == t) maxiou = fmaxf(maxiou, iou);
    if (valid && gj != t && sprc[gj] > 0.0f && iou > IOU_THR_C) sw += 1.0f;
  }
  float match = (valid && sgtc[t] > 0.0f) ? (1.0f - maxiou) : 0.0f;

  s_red[tid] = match;
  __syncthreads();
  if (tid < 32) {
    float tot = wmma_reduce256(s_red);
    if (tid == 0) atomicAdd(&ws[b * WS_PER_B + 5], tot);
  }
  __syncthreads();
  s_red[tid] = sw;
  __syncthreads();
  if (tid < 32) {
    float tot = wmma_reduce256(s_red);
    if (tid == 0) atomicAdd(&ws[b * WS_PER_B + 6], tot);
  }
}

// ---------------------------------------------------------------------------
// Kernel 3: temporal consistency. Thread t scans preds in original index
// order for track id t (== stable-sort grouping), accumulating |acc|.
// ---------------------------------------------------------------------------
__global__ void temporal_kernel(const float* __restrict__ pb, const int* __restrict__ pid,
                                float* __restrict__ ws) {
  const int b = blockIdx.x;
  const int tid = threadIdx.x;
  __shared__ float scx[NPRED], scy[NPRED];
  __shared__ int   spid[NPRED];
  __shared__ float s_red[TIDS];

  const float4* pb4 = (const float4*)pb;
  for (int i = tid; i < NPRED; i += blockDim.x) {
    float4 p = pb4[b * NPRED + i];
    scx[i] = (p.x + p.z) * 0.5f;
    scy[i] = (p.y + p.w) * 0.5f;
    spid[i] = pid[b * NPRED + i];
  }
  __syncthreads();

  float term = 0.0f;
  if (tid > 0) {
    float p1x = 0.0f, p1y = 0.0f, p2x = 0.0f, p2y = 0.0f, sum = 0.0f;
    int cnt = 0;
    for (int i = 0; i < NPRED; ++i) {       // LDS broadcast reads across lanes
      if (spid[i] == tid) {
        float cx = scx[i], cy = scy[i];
        if (cnt >= 2) {
          float ax = (cx - 2.0f * p1x + p2x) * scale_const();
          float ay = (cy - 2.0f * p1y + p2y) * scale_const();
          sum += sqrtf(ax * ax + ay * ay);
        }
        p2x = p1x; p2y = p1y; p1x = cx; p1y = cy;
        ++cnt;
      }
    }
    if (cnt > 2) term = sum / (float)(cnt - 2);
  }
  s_red[tid] = term;
  __syncthreads();
  if (tid < 32) {
    float tot = wmma_reduce256(s_red);
    if (tid == 0) {
      float nv = ws[b * WS_PER_B + 4];
      float nu = ws[b * WS_PER_B + 3];
      ws[b * WS_PER_B + 7] = (nv > 2.0f) ? tot / fmaxf(nu, 1.0f) : 0.0f;
    }
  }
}

// ---------------------------------------------------------------------------
// Kernel 4: combine 16 batches -> [total, tracking, spatial, temporal].
// ---------------------------------------------------------------------------
__global__ void finalize_kernel(const float* __restrict__ ws, float* __restrict__ out) {
  if (threadIdx.x != 0 || blockIdx.x != 0) return;
  float st = 0.0f, ss = 0.0f, sm = 0.0f, num = 0.0f;
  for (int b = 0; b < BATCH; ++b) {
    const float* w = ws + b * WS_PER_B;
    float fn = w[0], fp = w[1], sp = w[2], nu = w[3], nv = w[4];
    float mt = w[5], sw = w[6], tm = w[7];
    if (nv > 0.0f) {   // has-valid gate
      st += B_FN * fn + mt + G_SW * sw + B_FP * fp;
      ss += sp;
      sm += tm;
      num += nu;
    }
  }
  float norm = (num > 0.0f) ? num : 1.0f;
  float lt = st / norm, ls = ss / norm, lm = sm / norm;
  out[0] = A_TRK * lt + A_SPA * ls + A_TMP * lm;
  out[1] = lt;
  out[2] = ls;
  out[3] = lm;
}

// ---------------------------------------------------------------------------
extern "C" void kernel_launch(void* const* d_in, const int* in_sizes, int n_in,
                              void* d_out, int out_size, void* d_ws, size_t ws_size,
                              hipStream_t stream) {
  (void)in_sizes; (void)n_in; (void)out_size; (void)ws_size;
  const float* pb = (const float*)d_in[0];   // pred_boxes  [B,N,4] f32
  const float* gb = (const float*)d_in[1];   // gt_boxes    [B,M,4] f32
  const int*  pid = (const int*)d_in[2];     // pred ids    [B,N]   i32
  const int*  gid = (const int*)d_in[3];     // gt ids      [B,M]   i32
  float* ws = (float*)d_ws;                  // needs 16*520*4 = 33280 bytes
  float* out = (float*)d_out;                // 4 floats

  stats_kernel<<<BATCH, TIDS, 0, stream>>>(pb, gb, pid, gid, ws);
  iou_kernel<<<dim3(NPRED / TIDS, BATCH), TIDS, 0, stream>>>(pb, gb, pid, gid, ws);
  temporal_kernel<<<BATCH, TIDS, 0, stream>>>(pb, pid, ws);
  finalize_kernel<<<1, 32, 0, stream>>>(ws, out);
}